// dynamic_filter_channel_22771916603489
// MI455X (gfx1250) — compile-verified
//
#include <hip/hip_runtime.h>
#include <hip/hip_bf16.h>
#include <math.h>

// ---------------------------------------------------------------------------
// Problem constants (from reference): x (8,64,128,128) f32, K=3, GROUP=8,
// w_conv (72,64), w_gate (72,72), ln_gamma/beta (72).
// ---------------------------------------------------------------------------
#define N_   8
#define C_   64
#define H_   128
#define W_   128
#define CK_  72      // GROUP * K * K
#define HW_  (H_ * W_)
#define LN_EPS_ 1e-5f

typedef __attribute__((ext_vector_type(2))) float v2f;
typedef __attribute__((ext_vector_type(8))) float v8f;

// ---------------------------------------------------------------------------
// Kernel 1: global average pool. One block per (c, n); 256 threads, float4
// loads (16 per thread), LDS tree reduction.
// ---------------------------------------------------------------------------
__global__ void gap_kernel(const float* __restrict__ x, float* __restrict__ gap) {
    const int c = blockIdx.x;
    const int n = blockIdx.y;
    const float4* p4 = reinterpret_cast<const float4*>(
        x + ((size_t)n * C_ + c) * HW_);

    float s = 0.0f;
    #pragma unroll
    for (int it = 0; it < (HW_ / 4) / 256; ++it) {
        float4 v = p4[it * 256 + threadIdx.x];
        s += v.x + v.y + v.z + v.w;
    }

    __shared__ float red[256];
    red[threadIdx.x] = s;
    __syncthreads();
    for (int off = 128; off > 0; off >>= 1) {
        if (threadIdx.x < off) red[threadIdx.x] += red[threadIdx.x + off];
        __syncthreads();
    }
    if (threadIdx.x == 0)
        gap[n * C_ + c] = red[0] * (1.0f / (float)HW_);
}

// ---------------------------------------------------------------------------
// Kernel 2: dynamic-filter branch on ONE wave32 using V_WMMA_F32_16X16X4_F32.
//   lf  = gap @ w_conv^T            (8x64 @ 64x72) -> M=16(pad), N=5x16, K=16x4
//   lf *= sigmoid(lf @ w_gate^T)    (8x72 @ 72x72) -> K=18x4
//   LayerNorm(72), softmax over 9 taps per group -> filt (8,8,9)
//
// Out-of-tile operands use CLAMPED addresses + value select (v_cndmask), so
// no exec-mask branching around the loads and EXEC stays all-1s for WMMA.
//
// f32 WMMA fragment layout (ISA 7.12.2, 32-bit A 16x4):
//   lane m = lane&15, K-quadrant q = lane>>4; A vgpr0 = A[m][2q], vgpr1 = A[m][2q+1]
//   B (4x16) mirrored: vgpr0 = B[2q][nj], vgpr1 = B[2q+1][nj], nj = lane&15
//   C/D: vgpr r, lanes 0-15 -> row r, lanes 16-31 -> row r+8.
// ---------------------------------------------------------------------------
__global__ void filter_kernel(const float* __restrict__ gap,
                              const float* __restrict__ w_conv,
                              const float* __restrict__ w_gate,
                              const float* __restrict__ ln_gamma,
                              const float* __restrict__ ln_beta,
                              float* __restrict__ filt) {
    __shared__ float lf[N_][80];     // conv result   (8 x 72, padded stride)
    __shared__ float lf2[N_][80];    // gated result

    const int lane = threadIdx.x;        // 0..31
    const int m    = lane & 15;          // row in 16x16 tile
    const int q    = lane >> 4;          // K quadrant (0/1)

    const bool mOk = (m < N_);
    const int  mc  = mOk ? m : 0;        // clamped row (always valid address)

    // ---- stage 1: lf = gap @ w_conv^T via WMMA f32 16x16x4, K=64 ----
    for (int t = 0; t < 5; ++t) {        // N tiles: 5*16 = 80 >= 72
        v8f acc = {};
        const int  nj   = t * 16 + m;    // output column (filter feature idx)
        const bool njOk = (nj < CK_);
        const int  njc  = njOk ? nj : (CK_ - 1);
        #pragma unroll
        for (int kk = 0; kk < 16; ++kk) {  // K = 64 in steps of 4
            const int kb = kk * 4 + q * 2;
            float ax = gap[mc * C_ + kb];
            float ay = gap[mc * C_ + kb + 1];
            float bx = w_conv[njc * C_ + kb];
            float by = w_conv[njc * C_ + kb + 1];
            v2f a, b;
            a.x = mOk  ? ax : 0.0f;  a.y = mOk  ? ay : 0.0f;
            b.x = njOk ? bx : 0.0f;  b.y = njOk ? by : 0.0f;
            acc = __builtin_amdgcn_wmma_f32_16x16x4_f32(
                false, a, false, b, (short)0, acc, false, false);
        }
        if (q == 0 && nj < CK_) {        // lanes 0-15 hold rows 0..7 (vgpr r)
            #pragma unroll
            for (int r = 0; r < N_; ++r) lf[r][nj] = acc[r];
        }
        __syncthreads();
    }

    // ---- stage 2: g = lf @ w_gate^T (K=72); lf2 = lf * sigmoid(g) ----
    for (int t = 0; t < 5; ++t) {
        v8f acc = {};
        const int  nj   = t * 16 + m;
        const bool njOk = (nj < CK_);
        const int  njc  = njOk ? nj : (CK_ - 1);
        #pragma unroll
        for (int kk = 0; kk < 18; ++kk) {  // K = 72 in steps of 4
            const int kb = kk * 4 + q * 2;
            float ax = lf[mc][kb];         // LDS reads always in-bounds
            float ay = lf[mc][kb + 1];
            float bx = w_gate[njc * CK_ + kb];
            float by = w_gate[njc * CK_ + kb + 1];
            v2f a, b;
            a.x = mOk  ? ax : 0.0f;  a.y = mOk  ? ay : 0.0f;
            b.x = njOk ? bx : 0.0f;  b.y = njOk ? by : 0.0f;
            acc = __builtin_amdgcn_wmma_f32_16x16x4_f32(
                false, a, false, b, (short)0, acc, false, false);
        }
        if (q == 0 && nj < CK_) {
            #pragma unroll
            for (int r = 0; r < N_; ++r) {
                float g   = acc[r];
                float sig = 1.0f / (1.0f + __expf(-g));
                lf2[r][nj] = lf[r][nj] * sig;
            }
        }
        __syncthreads();
    }

    // ---- stage 3: LayerNorm(72) per sample (8 lanes, serial over 72) ----
    if (lane < N_) {
        float mu = 0.0f;
        for (int j = 0; j < CK_; ++j) mu += lf2[lane][j];
        mu *= (1.0f / (float)CK_);
        float var = 0.0f;
        for (int j = 0; j < CK_; ++j) {
            float d = lf2[lane][j] - mu;
            var += d * d;
        }
        var *= (1.0f / (float)CK_);
        const float inv = rsqrtf(var + LN_EPS_);
        for (int j = 0; j < CK_; ++j)
            lf2[lane][j] = (lf2[lane][j] - mu) * inv * ln_gamma[j] + ln_beta[j];
    }
    __syncthreads();

    // ---- stage 4: softmax over 9 taps per (sample, group): 64 groups ----
    for (int it = 0; it < 2; ++it) {
        const int gidx = it * 32 + lane;       // 0..63
        const int row  = gidx >> 3;            // sample
        const int grp  = gidx & 7;             // group
        float mx = -INFINITY;
        #pragma unroll
        for (int j = 0; j < 9; ++j) mx = fmaxf(mx, lf2[row][grp * 9 + j]);
        float e[9], s = 0.0f;
        #pragma unroll
        for (int j = 0; j < 9; ++j) { e[j] = __expf(lf2[row][grp * 9 + j] - mx); s += e[j]; }
        const float rs = 1.0f / s;
        #pragma unroll
        for (int j = 0; j < 9; ++j)
            filt[(row * 8 + grp) * 9 + j] = e[j] * rs;
    }
}

// ---------------------------------------------------------------------------
// Kernel 3: reflect-padded 3x3 weighted stencil (bandwidth-bound part).
// Block = 256 threads -> one 8x128 output tile of a single (n, c) plane.
// LDS halo tile 10 x 130 (stride 132). Filter coeffs are block-uniform.
// Each thread computes 4 CONSECUTIVE pixels and writes float4 (b128 stores).
// ---------------------------------------------------------------------------
__device__ __forceinline__ int reflect_(int p, int nmax) {
    // reflect mode without repeating edge: -1 -> 1, n -> n-2
    if (p < 0)      p = -p;
    if (p >= nmax)  p = 2 * nmax - 2 - p;
    return p;
}

__global__ void stencil_kernel(const float* __restrict__ x,
                               const float* __restrict__ filt,
                               float* __restrict__ low,
                               float* __restrict__ high) {
    __shared__ float tile[10][132];

    const int n  = blockIdx.z;
    const int c  = blockIdx.y;
    const int h0 = blockIdx.x * 8;
    const float* xc = x + ((size_t)n * C_ + c) * HW_;

    // per-(n, group) 3x3 filter: uniform across the block
    const float* f = filt + ((size_t)(n * 8 + (c >> 3))) * 9;
    const float f0 = f[0], f1 = f[1], f2 = f[2];
    const float f3 = f[3], f4 = f[4], f5 = f[5];
    const float f6 = f[6], f7 = f[7], f8 = f[8];

    // cooperative halo load with reflect padding (10 x 130 elements)
    for (int i = threadIdx.x; i < 10 * 130; i += 256) {
        const int r  = i / 130;
        const int cc = i - r * 130;
        const int gh = reflect_(h0 + r - 1, H_);
        const int gw = reflect_(cc - 1, W_);
        tile[r][cc] = xc[gh * W_ + gw];
    }
    __syncthreads();

    // 4 consecutive pixels per thread -> float4 stores
    const int i = threadIdx.x * 4;          // 0..1020
    const int r = i >> 7;                   // 0..7
    const int w = i & 127;                  // multiple of 4

    float4 lo, hi;
    float* lop = &lo.x;
    float* hip_ = &hi.x;
    #pragma unroll
    for (int j = 0; j < 4; ++j) {
        const int wc = w + j;
        const float acc =
            f0 * tile[r    ][wc] + f1 * tile[r    ][wc + 1] + f2 * tile[r    ][wc + 2] +
            f3 * tile[r + 1][wc] + f4 * tile[r + 1][wc + 1] + f5 * tile[r + 1][wc + 2] +
            f6 * tile[r + 2][wc] + f7 * tile[r + 2][wc + 1] + f8 * tile[r + 2][wc + 2];
        lop[j]  = acc;
        hip_[j] = tile[r + 1][wc + 1] - acc;
    }

    const size_t o = ((size_t)n * C_ + c) * HW_ + (size_t)(h0 + r) * W_ + w;
    *reinterpret_cast<float4*>(low  + o) = lo;
    *reinterpret_cast<float4*>(high + o) = hi;
}

// ---------------------------------------------------------------------------
// Host-side launcher
// ---------------------------------------------------------------------------
extern "C" void kernel_launch(void* const* d_in, const int* in_sizes, int n_in,
                              void* d_out, int out_size, void* d_ws, size_t ws_size,
                              hipStream_t stream) {
    const float* x        = (const float*)d_in[0];
    const float* w_conv   = (const float*)d_in[1];
    const float* w_gate   = (const float*)d_in[2];
    const float* ln_gamma = (const float*)d_in[3];
    const float* ln_beta  = (const float*)d_in[4];

    float* out  = (float*)d_out;
    float* gap  = (float*)d_ws;          // 512 floats
    float* filt = gap + N_ * C_;         // 576 floats

    const size_t plane = (size_t)N_ * C_ * HW_;   // 8,388,608

    gap_kernel<<<dim3(C_, N_), 256, 0, stream>>>(x, gap);
    filter_kernel<<<1, 32, 0, stream>>>(gap, w_conv, w_gate, ln_gamma, ln_beta, filt);
    stencil_kernel<<<dim3(H_ / 8, C_, N_), 256, 0, stream>>>(
        x, filt, out, out + plane);
}